// DataEmbedding_cycle_pos_90271622627788
// MI455X (gfx1250) — compile-verified
//
#include <hip/hip_runtime.h>
#include <math.h>

typedef float v2f __attribute__((ext_vector_type(2)));
typedef float v8f __attribute__((ext_vector_type(8)));

#define NB   16      // batch
#define TT   512     // time steps
#define NCH  32      // input channels n
#define DM   512     // d_model
#define KTOT 96      // GEMM K = 3 taps * 32 channels
#define BT   (NB*TT) // GEMM M = 8192

// LDS strides (floats). Even => 8B-aligned b64 loads; odd multiples of rows
// spread across the 64 banks (34 mod 64 and 98 mod 64 give distinct banks for
// 16 consecutive rows).
#define XS_STRIDE 34
#define BS_STRIDE 98

// ---------------------------------------------------------------------------
// K0: build the shared sinusoidal table T[512][512].
// All four "fixed" embedding tables and the cycle table use the identical
// formula; every row index used anywhere is < 512, so one table suffices.
// ---------------------------------------------------------------------------
__global__ __launch_bounds__(256)
void build_table_kernel(float* __restrict__ T) {
  int idx = blockIdx.x * blockDim.x + threadIdx.x;   // 0 .. 512*512-1
  int r = idx >> 9;
  int d = idx & (DM - 1);
  int j2 = d & ~1;                                   // 2*floor(d/2)
  float div = expf(-(float)j2 * (logf(10000.0f) / (float)DM));
  float a = (float)r * div;
  T[idx] = (d & 1) ? cosf(a) : sinf(a);
}

// ---------------------------------------------------------------------------
// K1: per-(b,n) series -> dominant rFFT bin -> period.
// Brute-force DFT over 257 bins; one workgroup (256 threads) per series.
// top_k(k=1) tie-break = lowest index.
// ---------------------------------------------------------------------------
__global__ __launch_bounds__(256)
void period_kernel(const float* __restrict__ x, float* __restrict__ periods) {
  __shared__ float s[TT];
  __shared__ float smag[256];
  __shared__ int   sidx[256];
  int wg  = blockIdx.x;            // b*32 + n
  int b   = wg >> 5, n = wg & 31;
  int tid = threadIdx.x;

  for (int u = tid; u < TT; u += 256)
    s[u] = x[((b * TT) + u) * NCH + n];
  __syncthreads();

  float bestm = -1.0f; int besti = 0;
  for (int f = tid; f <= TT / 2; f += 256) {         // thread 0 also does bin 256
    float w = -6.28318530717958647692f * (float)f / (float)TT;
    float re = 0.f, im = 0.f;
    for (int u = 0; u < TT; ++u) {
      float ang = w * (float)u;
      re += s[u] * __cosf(ang);
      im += s[u] * __sinf(ang);
    }
    float m = re * re + im * im;                     // |X|^2: same argmax as |X|
    if (m > bestm || (m == bestm && f < besti)) { bestm = m; besti = f; }
  }
  smag[tid] = bestm; sidx[tid] = besti;
  __syncthreads();
  for (int sft = 128; sft > 0; sft >>= 1) {
    if (tid < sft) {
      float om = smag[tid + sft]; int oi = sidx[tid + sft];
      if (om > smag[tid] || (om == smag[tid] && oi < sidx[tid])) {
        smag[tid] = om; sidx[tid] = oi;
      }
    }
    __syncthreads();
  }
  if (tid == 0) {
    int idx = sidx[0];
    // fftfreq(512)[idx]: idx/512 for idx<256, -0.5 at the Nyquist bin.
    float freq = (idx == TT / 2) ? -0.5f : ((float)idx / (float)TT);
    float p = (float)TT / freq;                      // idx==0 -> +inf -> clipped
    p = fminf(fmaxf(p, 1.0f), (float)TT);
    periods[wg] = p;
  }
}

// ---------------------------------------------------------------------------
// K2: circular conv1d as a GEMM via V_WMMA_F32_16X16X4_F32.
// Workgroup tile: 128(M) x 64(N); 8 waves, each wave a 16x64 strip with
// 4 f32 16x16 accumulators; K loop = 24 steps of 4.
// A[r, tap*32+chan] = x[b, (t0-1+r+tap) mod 512, chan]  (circular pad)
// B[c, d]           = conv_w[d, c%32, c/32]             (staged transposed)
// ---------------------------------------------------------------------------
__global__ __launch_bounds__(256)
void conv_wmma_kernel(const float* __restrict__ x, const float* __restrict__ w,
                      float* __restrict__ out) {
  __shared__ float xs[130 * XS_STRIDE];   // times t0-1 .. t0+128, 32 channels
  __shared__ float bs[64 * BS_STRIDE];    // bs[n_local][c], c = tap*32+chan

  int mt = blockIdx.x;                    // 0..63  (M tiles of 128 rows)
  int nt = blockIdx.y;                    // 0..7   (N tiles of 64 cols)
  int b  = mt >> 2;                       // 128 | 512, tiles never cross batch
  int t0 = (mt & 3) * 128;
  int n0 = nt * 64;
  int tid = threadIdx.x;

  // Stage the circular x window (coalesced over channels).
  for (int f = tid; f < 130 * NCH; f += 256) {
    int r = f >> 5, i = f & 31;
    int ttw = (t0 - 1 + r) & (TT - 1);
    xs[r * XS_STRIDE + i] = x[((b * TT) + ttw) * NCH + i];
  }
  // Stage B transposed: bs[n][c] = conv_w[(n0+n)*96 + chan*3 + tap].
  for (int f = tid; f < 64 * KTOT; f += 256) {
    int nl = f / KTOT, c = f - nl * KTOT;
    int tap = c >> 5, chan = c & 31;
    bs[nl * BS_STRIDE + c] = w[(n0 + nl) * KTOT + chan * 3 + tap];
  }
  __syncthreads();

  int lane = tid & 31;
  int wave = tid >> 5;
  int half = lane >> 4;                   // 0: K pair 0/1, 1: K pair 2/3
  int l16  = lane & 15;
  int arow = wave * 16 + l16;             // A-matrix M row within tile

  v8f acc0 = {}, acc1 = {}, acc2 = {}, acc3 = {};

  for (int k0 = 0; k0 < KTOT; k0 += 4) {
    int c   = k0 + half * 2;              // even => chan,chan+1 in same tap
    int tap = c >> 5, chan = c & 31;
    // A 16x4 f32 layout: VGPR0 = K{0,2}, VGPR1 = K{1,3} across lane halves.
    v2f a  = *(const v2f*)&xs[(arow + tap) * XS_STRIDE + chan];
    // B 4x16 f32 layout (symmetric): lane = N column, VGPR pair = K pair.
    v2f b0 = *(const v2f*)&bs[(l16 +  0) * BS_STRIDE + c];
    v2f b1 = *(const v2f*)&bs[(l16 + 16) * BS_STRIDE + c];
    v2f b2 = *(const v2f*)&bs[(l16 + 32) * BS_STRIDE + c];
    v2f b3 = *(const v2f*)&bs[(l16 + 48) * BS_STRIDE + c];
    acc0 = __builtin_amdgcn_wmma_f32_16x16x4_f32(false, a, false, b0, (short)0, acc0, false, false);
    acc1 = __builtin_amdgcn_wmma_f32_16x16x4_f32(false, a, false, b1, (short)0, acc1, false, false);
    acc2 = __builtin_amdgcn_wmma_f32_16x16x4_f32(false, a, false, b2, (short)0, acc2, false, false);
    acc3 = __builtin_amdgcn_wmma_f32_16x16x4_f32(false, a, false, b3, (short)0, acc3, false, false);
  }

  // C/D layout: VGPR v -> M = v + 8*(lane/16), N = lane%16.
  int col0  = n0 + l16;
  int rbase = b * TT * 0 + (b * TT) + t0 + wave * 16 + half * 8;
#pragma unroll
  for (int v = 0; v < 8; ++v) {
    int row = rbase + v;
    out[row * DM + col0 +  0] = acc0[v];
    out[row * DM + col0 + 16] = acc1[v];
    out[row * DM + col0 + 32] = acc2[v];
    out[row * DM + col0 + 48] = acc3[v];
  }
}

// ---------------------------------------------------------------------------
// K3: out += temporal (4 table rows via x_mark) + cycle (mean of 32 rows via
// pidx = fmod(t, period)). One workgroup per (b,t) output row; table rows are
// L2-resident (1 MB table, 192 MB L2), reads coalesced over d.
// ---------------------------------------------------------------------------
__global__ __launch_bounds__(256)
void epilogue_kernel(const int* __restrict__ xmark, const float* __restrict__ T,
                     const float* __restrict__ periods, float* __restrict__ out) {
  __shared__ int prow[NCH];
  __shared__ int mrow[4];
  int wg  = blockIdx.x;                    // b*512 + t
  int b   = wg >> 9;
  int t   = wg & (TT - 1);
  int tid = threadIdx.x;

  if (tid < NCH) {
    float p = periods[b * NCH + tid];
    prow[tid] = (int)fmodf((float)t, p);   // < 512 always
  }
  if (tid >= 32 && tid < 36)
    mrow[tid - 32] = xmark[wg * 4 + (tid - 32)];   // values 0..6
  __syncthreads();

  for (int d = tid; d < DM; d += 256) {
    float s = T[mrow[0] * DM + d] + T[mrow[1] * DM + d] +
              T[mrow[2] * DM + d] + T[mrow[3] * DM + d];
    float c = 0.f;
#pragma unroll
    for (int i = 0; i < NCH; ++i) c += T[prow[i] * DM + d];
    out[wg * DM + d] += s + c * (1.0f / (float)NCH);
  }
}

// ---------------------------------------------------------------------------
extern "C" void kernel_launch(void* const* d_in, const int* in_sizes, int n_in,
                              void* d_out, int out_size, void* d_ws, size_t ws_size,
                              hipStream_t stream) {
  const float* x     = (const float*)d_in[0];   // (16,512,32) f32
  const int*   xmark = (const int*)  d_in[1];   // (16,512,4)  i32
  const float* cw    = (const float*)d_in[2];   // (512,32,3)  f32
  float* out = (float*)d_out;                   // (16,512,512) f32

  float* T       = (float*)d_ws;                // 512*512 floats (1 MB)
  float* periods = T + TT * DM;                 // 512 floats

  build_table_kernel<<<(TT * DM) / 256, 256, 0, stream>>>(T);
  period_kernel<<<NB * NCH, 256, 0, stream>>>(x, periods);

  dim3 g2(BT / 128, DM / 64);
  conv_wmma_kernel<<<g2, 256, 0, stream>>>(x, cw, out);

  epilogue_kernel<<<BT, 256, 0, stream>>>(xmark, T, periods, out);
}